// LightGCN_4698694222366
// MI455X (gfx1250) — compile-verified
//
#include <hip/hip_runtime.h>

// LightGCN forward for MI455X (gfx1250, wave32).
//
// Strategy: node-feature buffers (N*64*4 = 25.6 MB) are L2-resident (192 MB L2),
// so propagation runs as L2-bound gather/atomic-scatter. Scoring uses
// V_WMMA_F32_16X16X4_F32 chains (exact f32 math), one wave per 16 label pairs,
// extracting the diagonal of A(16x64) x B(64x16).

typedef float v2f __attribute__((ext_vector_type(2)));
typedef float v8f __attribute__((ext_vector_type(8)));

// ---------------- elementwise helpers (float4 vectorized) ----------------

__global__ void zero_f32v4(float* __restrict__ p, long long n4) {
    long long i = blockIdx.x * (long long)blockDim.x + threadIdx.x;
    if (i < n4) ((float4*)p)[i] = make_float4(0.f, 0.f, 0.f, 0.f);
}

__global__ void scale_f32v4(const float* __restrict__ src, float* __restrict__ dst,
                            float a, long long n4) {
    long long i = blockIdx.x * (long long)blockDim.x + threadIdx.x;
    if (i < n4) {
        float4 v = ((const float4*)src)[i];
        ((float4*)dst)[i] = make_float4(a * v.x, a * v.y, a * v.z, a * v.w);
    }
}

__global__ void axpy_f32v4(const float* __restrict__ x, float* __restrict__ y,
                           float a, long long n4) {
    long long i = blockIdx.x * (long long)blockDim.x + threadIdx.x;
    if (i < n4) {
        float4 v = ((const float4*)x)[i];
        float4 o = ((float4*)y)[i];
        o.x += a * v.x; o.y += a * v.y; o.z += a * v.z; o.w += a * v.w;
        ((float4*)y)[i] = o;
    }
}

// ---------------- degree / norm ----------------

__global__ void degree_kernel(const int* __restrict__ col, float* __restrict__ deg,
                              long long E) {
    long long e = blockIdx.x * (long long)blockDim.x + threadIdx.x;
    if (e < E) atomicAdd(deg + col[e], 1.0f);
}

__global__ void dinv_kernel(float* __restrict__ deg, long long N) {
    long long i = blockIdx.x * (long long)blockDim.x + threadIdx.x;
    if (i < N) {
        float d = deg[i];
        deg[i] = (d > 0.f) ? rsqrtf(d) : 0.f;   // v_rsq_f32
    }
}

__global__ void norm_kernel(const int* __restrict__ row, const int* __restrict__ col,
                            const float* __restrict__ dinv, float* __restrict__ nrm,
                            long long E) {
    long long e = blockIdx.x * (long long)blockDim.x + threadIdx.x;
    if (e < E) nrm[e] = dinv[row[e]] * dinv[col[e]];
}

// ---------------- propagation: 16 threads per edge, float4 per thread ----------------
// gather x[row] (coalesced 256B row), scale by norm, atomic-scatter to xn[col].
// xn fits in L2, so atomics resolve at the L2 atomic units.

__global__ void scatter_edges(const float* __restrict__ x, float* __restrict__ xn,
                              const float* __restrict__ nrm,
                              const int* __restrict__ row, const int* __restrict__ col,
                              long long E) {
    long long tid = blockIdx.x * (long long)blockDim.x + threadIdx.x;
    long long e = tid >> 4;          // 16 threads cooperate on one edge
    int sub = (int)(tid & 15);       // each handles 4 consecutive dims
    if (e >= E) return;
    int r = row[e];
    int c = col[e];
    float w = nrm[e];
    const float4 v = *(const float4*)(x + (size_t)r * 64 + sub * 4);
    float* dst = xn + (size_t)c * 64 + sub * 4;
    atomicAdd(dst + 0, w * v.x);
    atomicAdd(dst + 1, w * v.y);
    atomicAdd(dst + 2, w * v.z);
    atomicAdd(dst + 3, w * v.w);
}

// ---------------- scoring via V_WMMA_F32_16X16X4_F32 ----------------
// One wave -> 16 label pairs. D(m,n) = dot(out[src_m], out[dst_n]); keep diagonal.
// A 16x4 f32 layout: lane L holds M=L&15, K = {2*(L>>4), 2*(L>>4)+1} in vgpr{0,1}.
// B 4x16 f32 layout is symmetric with N = L&15. Chain C across 16 K-chunks.

__global__ void score_wmma(const float* __restrict__ outb,
                           const int* __restrict__ lsrc, const int* __restrict__ ldst,
                           float* __restrict__ scores, long long L) {
    long long wave = (blockIdx.x * (long long)blockDim.x + threadIdx.x) >> 5;
    int lane = threadIdx.x & 31;
    long long base = wave * 16;
    if (base >= L) return;                       // wave-uniform: EXEC stays all-1 for WMMA

    int m = lane & 15;
    int hi = lane >> 4;                          // selects K pair {0,1} vs {2,3}
    long long li = base + m;
    if (li >= L) li = L - 1;                     // safe clamp (L%16==0 in practice)
    const float* srow = outb + (size_t)lsrc[li] * 64;
    const float* drow = outb + (size_t)ldst[li] * 64;

    v8f c = {};
#pragma unroll
    for (int k = 0; k < 64; k += 4) {
        int off = k + 2 * hi;
        v2f a = *(const v2f*)(srow + off);       // A[M=m][k+2hi .. +1]
        v2f b = *(const v2f*)(drow + off);       // B[k+2hi .. +1][N=m]
        c = __builtin_amdgcn_wmma_f32_16x16x4_f32(
                /*neg_a=*/false, a, /*neg_b=*/false, b,
                /*c_mod=*/(short)0, c, /*reuse_a=*/false, /*reuse_b=*/false);
    }

    // Diagonal D(t,t): t<8 -> lane t, vgpr t;  t>=8 -> lane t+16, vgpr t-8.
    int sel = -1;
    long long widx = -1;
    if (lane < 8)        { sel = lane;      widx = base + lane; }
    else if (lane >= 24) { sel = lane - 24; widx = base + (lane - 16); }
    if (sel >= 0 && widx < L) {
        float val = 0.f;
#pragma unroll
        for (int r = 0; r < 8; ++r)
            if (sel == r) val = c[r];
        scores[widx] = val;
    }
}

// ---------------- host orchestration ----------------

extern "C" void kernel_launch(void* const* d_in, const int* in_sizes, int n_in,
                              void* d_out, int out_size, void* d_ws, size_t ws_size,
                              hipStream_t stream) {
    (void)n_in; (void)out_size; (void)ws_size;

    const float* emb = (const float*)d_in[0];
    const int*   ei  = (const int*)d_in[1];   // [2, E] flat (int32 under default JAX x64-off)
    const int*   eli = (const int*)d_in[2];   // [2, L] flat

    const long long N = in_sizes[0] / 64;
    const long long E = in_sizes[1] / 2;
    const long long L = in_sizes[2] / 2;

    const int* row  = ei;          // src
    const int* colv = ei + E;      // dst
    const int* lsrc = eli;
    const int* ldst = eli + L;

    // Workspace layout (256B aligned): deg/dinv | norm | bufA | bufB | out
    char* ws = (char*)d_ws;
    size_t o = 0;
    auto al = [](size_t x) { return (x + 255) & ~(size_t)255; };
    float* deg  = (float*)(ws + o); o = al(o + (size_t)N * 4);
    float* nrm  = (float*)(ws + o); o = al(o + (size_t)E * 4);
    float* bufA = (float*)(ws + o); o = al(o + (size_t)N * 64 * 4);
    float* bufB = (float*)(ws + o); o = al(o + (size_t)N * 64 * 4);
    float* outb = (float*)(ws + o); o = al(o + (size_t)N * 64 * 4);

    const float alpha = 1.0f / 4.0f;   // 1/(NUM_LAYERS+1), NUM_LAYERS=3
    dim3 blk(256);
    auto nb = [](long long t) { return dim3((unsigned)((t + 255) / 256)); };

    const long long n4 = N * 64 / 4;

    // 1) degree -> d^{-1/2} -> per-edge norm
    zero_f32v4<<<nb((N + 3) / 4), blk, 0, stream>>>(deg, (N + 3) / 4);
    degree_kernel<<<nb(E), blk, 0, stream>>>(colv, deg, E);
    dinv_kernel<<<nb(N), blk, 0, stream>>>(deg, N);
    norm_kernel<<<nb(E), blk, 0, stream>>>(row, colv, deg, nrm, E);

    // 2) out = alpha * emb
    scale_f32v4<<<nb(n4), blk, 0, stream>>>(emb, outb, alpha, n4);

    // 3) three LGConv layers, ping-pong bufA/bufB (emb never written)
    const float* cur = emb;
    float* pp[2] = { bufA, bufB };
    for (int l = 0; l < 3; ++l) {
        float* tgt = pp[l & 1];
        zero_f32v4<<<nb(n4), blk, 0, stream>>>(tgt, n4);
        scatter_edges<<<nb(E * 16), blk, 0, stream>>>(cur, tgt, nrm, row, colv, E);
        axpy_f32v4<<<nb(n4), blk, 0, stream>>>(tgt, outb, alpha, n4);
        cur = tgt;
    }

    // 4) WMMA scoring: one wave per 16 label pairs
    const long long tiles = (L + 15) / 16;
    score_wmma<<<nb(tiles * 32), blk, 0, stream>>>(outb, lsrc, ldst, (float*)d_out, L);
}